// Classifier_40810779246751
// MI455X (gfx1250) — compile-verified
//
#include <hip/hip_runtime.h>
#include <hip/hip_bf16.h>
#include <math.h>

#define N_NODES 100000
#define D_IN    128
#define H_HEADS 8
#define C_CH    8
#define HC      64
#define NOUT_C  2
#define E_EDGES 1600000

typedef __attribute__((ext_vector_type(2))) float v2f;
typedef __attribute__((ext_vector_type(8))) float v8f;

__device__ __forceinline__ unsigned f2ord(float f) {
    unsigned u = __float_as_uint(f);
    return (u & 0x80000000u) ? ~u : (u | 0x80000000u);
}
__device__ __forceinline__ float ord2f(unsigned u) {
    return __uint_as_float((u & 0x80000000u) ? (u & 0x7FFFFFFFu) : ~u);
}

// -------------------------------------------------------------------------
// Kernel 1: h = x @ W  via V_WMMA_F32_16X16X4_F32 (wave32, 16x16 tile/wave)
// blockDim = 128 (4 waves); wave w handles column tile w (HC = 64 = 4*16).
// gridDim.x = N/16 = 6250 row tiles (exact, no tail -> EXEC all ones).
// -------------------------------------------------------------------------
__global__ __launch_bounds__(128) void gemm_xw_wmma(const float* __restrict__ x,
                                                    const float* __restrict__ W,
                                                    float* __restrict__ h) {
    const int wave    = threadIdx.x >> 5;     // 0..3  -> column tile
    const int lane    = threadIdx.x & 31;
    const int half    = lane >> 4;            // 0: K pair {0,1}, 1: K pair {2,3}
    const int l16     = lane & 15;
    const int rowBase = blockIdx.x * 16;
    const int colBase = wave * 16;

    const float* xrow = x + (size_t)(rowBase + l16) * D_IN;
    v8f c = {};
    #pragma unroll 4
    for (int kk = 0; kk < D_IN; kk += 4) {
        const int k0 = kk + 2 * half;
        v2f a, b;
        // A (16x4): lane -> row l16, K = k0, k0+1
        a.x = xrow[k0];
        a.y = xrow[k0 + 1];
        // B (4x16): lane -> col colBase+l16, K rows k0, k0+1
        b.x = W[(size_t)(k0    ) * HC + colBase + l16];
        b.y = W[(size_t)(k0 + 1) * HC + colBase + l16];
        c = __builtin_amdgcn_wmma_f32_16x16x4_f32(false, a, false, b,
                                                  (short)0, c, false, false);
    }
    // D (16x16 f32): VGPR r -> row r (lanes 0-15), row r+8 (lanes 16-31)
    float* hout = h + colBase + l16;
    #pragma unroll
    for (int r = 0; r < 8; ++r) {
        hout[(size_t)(rowBase + r + 8 * half) * HC] = c[r];
    }
}

// -------------------------------------------------------------------------
// Kernel 2: workspace init.  m -> ord(-inf), denom -> 0, accum -> 0
// -------------------------------------------------------------------------
__global__ void init_ws(unsigned* __restrict__ mord,
                        float* __restrict__ denom,
                        float* __restrict__ accum) {
    const int i = blockIdx.x * blockDim.x + threadIdx.x;
    if (i < N_NODES * H_HEADS) {
        mord[i]  = 0x007FFFFFu;   // f2ord(-inf)
        denom[i] = 0.0f;
    }
    if (i < N_NODES * HC) accum[i] = 0.0f;
}

// -------------------------------------------------------------------------
// Kernel 3: per-(node,head) attention scores
// -------------------------------------------------------------------------
__global__ void att_scores(const float* __restrict__ h,
                           const float* __restrict__ att_src,
                           const float* __restrict__ att_dst,
                           float* __restrict__ a_src,
                           float* __restrict__ a_dst) {
    const int i = blockIdx.x * blockDim.x + threadIdx.x;   // n*H + head
    if (i >= N_NODES * H_HEADS) return;
    const int n  = i >> 3;
    const int hh = i & 7;
    const float* hp = h + (size_t)n * HC + hh * C_CH;
    const float* as = att_src + hh * C_CH;
    const float* ad = att_dst + hh * C_CH;
    float s = 0.0f, d = 0.0f;
    #pragma unroll
    for (int cc = 0; cc < C_CH; ++cc) {
        const float v = hp[cc];
        s += v * as[cc];
        d += v * ad[cc];
    }
    a_src[i] = s;
    a_dst[i] = d;
}

__device__ __forceinline__ float leaky02(float v) {
    return v > 0.0f ? v : 0.2f * v;
}

// -------------------------------------------------------------------------
// Kernel 4: segment max over dst (order-preserving uint atomicMax)
// one thread per (edge, head)
// -------------------------------------------------------------------------
__global__ void edge_max(const long long* __restrict__ src,
                         const long long* __restrict__ dst,
                         const float* __restrict__ a_src,
                         const float* __restrict__ a_dst,
                         unsigned* __restrict__ mord) {
    const long long i = (long long)blockIdx.x * blockDim.x + threadIdx.x;
    if (i >= (long long)E_EDGES * H_HEADS) return;
    const long long e = i >> 3;
    const int hh = (int)(i & 7);
    const long long s = src[e], d = dst[e];
    const float ev = leaky02(a_src[s * H_HEADS + hh] + a_dst[d * H_HEADS + hh]);
    atomicMax(&mord[d * H_HEADS + hh], f2ord(ev));
}

// -------------------------------------------------------------------------
// Kernel 5: m = isfinite(max) ? max : 0   (in-place reinterpret)
// -------------------------------------------------------------------------
__global__ void finalize_m(unsigned* __restrict__ mord) {
    const int i = blockIdx.x * blockDim.x + threadIdx.x;
    if (i >= N_NODES * H_HEADS) return;
    float m = ord2f(mord[i]);
    if (!__builtin_isfinite(m)) m = 0.0f;
    ((float*)mord)[i] = m;
}

// -------------------------------------------------------------------------
// Kernel 6: denom[dst,h] += exp(e - m[dst,h])
// -------------------------------------------------------------------------
__global__ void edge_expsum(const long long* __restrict__ src,
                            const long long* __restrict__ dst,
                            const float* __restrict__ a_src,
                            const float* __restrict__ a_dst,
                            const float* __restrict__ m,
                            float* __restrict__ denom) {
    const long long i = (long long)blockIdx.x * blockDim.x + threadIdx.x;
    if (i >= (long long)E_EDGES * H_HEADS) return;
    const long long e = i >> 3;
    const int hh = (int)(i & 7);
    const long long s = src[e], d = dst[e];
    const float ev = leaky02(a_src[s * H_HEADS + hh] + a_dst[d * H_HEADS + hh]);
    const float ex = __expf(ev - m[d * H_HEADS + hh]);
    atomicAdd(&denom[d * H_HEADS + hh], ex);
}

// -------------------------------------------------------------------------
// Kernel 7: accum[dst, h*8+c] += h[src, h*8+c] * alpha(e,h)
// -------------------------------------------------------------------------
__global__ void edge_aggregate(const long long* __restrict__ src,
                               const long long* __restrict__ dst,
                               const float* __restrict__ a_src,
                               const float* __restrict__ a_dst,
                               const float* __restrict__ m,
                               const float* __restrict__ denom,
                               const float* __restrict__ h,
                               float* __restrict__ accum) {
    const long long i = (long long)blockIdx.x * blockDim.x + threadIdx.x;
    if (i >= (long long)E_EDGES * H_HEADS) return;
    const long long e = i >> 3;
    const int hh = (int)(i & 7);
    const long long s = src[e], d = dst[e];
    const float ev    = leaky02(a_src[s * H_HEADS + hh] + a_dst[d * H_HEADS + hh]);
    const float ex    = __expf(ev - m[d * H_HEADS + hh]);
    const float alpha = ex / denom[d * H_HEADS + hh];
    const float* hs = h     + s * HC + hh * C_CH;
    float*       ac = accum + d * HC + hh * C_CH;
    #pragma unroll
    for (int cc = 0; cc < C_CH; ++cc) {
        atomicAdd(&ac[cc], hs[cc] * alpha);
    }
}

// -------------------------------------------------------------------------
// Kernel 8: bias + relu + dropout(x2) + [64x2] classifier
// -------------------------------------------------------------------------
__global__ void head_out(const float* __restrict__ accum,
                         const float* __restrict__ bias_gat,
                         const unsigned char* __restrict__ mask,
                         const float* __restrict__ W_lin,
                         const float* __restrict__ b_lin,
                         float* __restrict__ out) {
    const int n = blockIdx.x * blockDim.x + threadIdx.x;
    if (n >= N_NODES) return;
    float o0 = b_lin[0], o1 = b_lin[1];
    const float* ar = accum + (size_t)n * HC;
    const unsigned char* mr = mask + (size_t)n * HC;
    #pragma unroll 8
    for (int j = 0; j < HC; ++j) {
        float v = ar[j] + bias_gat[j];
        v = v > 0.0f ? v : 0.0f;
        v = mr[j] ? v * 2.0f : 0.0f;
        o0 += v * W_lin[j * NOUT_C + 0];
        o1 += v * W_lin[j * NOUT_C + 1];
    }
    out[n * NOUT_C + 0] = o0;
    out[n * NOUT_C + 1] = o1;
}

// -------------------------------------------------------------------------
extern "C" void kernel_launch(void* const* d_in, const int* in_sizes, int n_in,
                              void* d_out, int out_size, void* d_ws, size_t ws_size,
                              hipStream_t stream) {
    const float*         x       = (const float*)d_in[0];
    const long long*     eidx    = (const long long*)d_in[1];   // [2, E]
    const unsigned char* dmask   = (const unsigned char*)d_in[2];
    const float*         W       = (const float*)d_in[3];
    const float*         att_src = (const float*)d_in[4];
    const float*         att_dst = (const float*)d_in[5];
    const float*         bias_g  = (const float*)d_in[6];
    const float*         W_lin   = (const float*)d_in[7];
    const float*         b_lin   = (const float*)d_in[8];
    float*               out     = (float*)d_out;

    const long long* src = eidx;
    const long long* dst = eidx + E_EDGES;

    // workspace layout (floats): h | a_src | a_dst | m/mord | denom | accum
    float*    h     = (float*)d_ws;
    float*    a_src = h + (size_t)N_NODES * HC;
    float*    a_dst = a_src + (size_t)N_NODES * H_HEADS;
    unsigned* mord  = (unsigned*)(a_dst + (size_t)N_NODES * H_HEADS);
    float*    m     = (float*)mord;                      // reused in place
    float*    denom = (float*)(mord + (size_t)N_NODES * H_HEADS);
    float*    accum = denom + (size_t)N_NODES * H_HEADS;

    const int T = 256;
    const int nh_blocks   = (N_NODES * H_HEADS + T - 1) / T;
    const int nhc_blocks  = (N_NODES * HC + T - 1) / T;
    const int eh_blocks   = (int)(((long long)E_EDGES * H_HEADS + T - 1) / T);
    const int node_blocks = (N_NODES + T - 1) / T;

    init_ws<<<nhc_blocks, T, 0, stream>>>(mord, denom, accum);
    gemm_xw_wmma<<<N_NODES / 16, 128, 0, stream>>>(x, W, h);
    att_scores<<<nh_blocks, T, 0, stream>>>(h, att_src, att_dst, a_src, a_dst);
    edge_max<<<eh_blocks, T, 0, stream>>>(src, dst, a_src, a_dst, mord);
    finalize_m<<<nh_blocks, T, 0, stream>>>(mord);
    edge_expsum<<<eh_blocks, T, 0, stream>>>(src, dst, a_src, a_dst, m, denom);
    edge_aggregate<<<eh_blocks, T, 0, stream>>>(src, dst, a_src, a_dst, m, denom, h, accum);
    head_out<<<node_blocks, T, 0, stream>>>(accum, bias_g, dmask, W_lin, b_lin, out);
}